// GraphRecurrent_22110491640095
// MI455X (gfx1250) — compile-verified
//
#include <hip/hip_runtime.h>
#include <math.h>

#define N_NODES 50000
#define N_EDGES 600000
#define H 128

typedef __attribute__((ext_vector_type(2))) float v2f;
typedef __attribute__((ext_vector_type(8))) float v8f;

// ---- workspace layout (float offsets) ----
#define OFF_U   0          // 512  : W_ih @ nn1_W
#define OFF_V   512        // 512  : W_ih @ nn1_b + b_ih + b_hh
#define OFF_P   1024       // 128  : lin3_W @ edge_map_W
#define OFF_Q   1152       // 128  : lin3_W @ edge_map_b + lin3_b
#define OFF_SC  1280       // 2    : alpha, beta
#define OFF_S   2048       // N    : conv1 scalar aggregation (atomic)
#define OFF_A   52048      // N    : dec_W[:128] . h3[i]
#define OFF_B   102048     // N    : dec_W[128:] . h3[i]
#define OFF_H2  152048     // N*H  : LSTM output
#define OFF_AG  6552048    // N*H  : conv3 aggregation (atomic)
#define OFF_H3  12952048   // N*H  : conv3 output
// total = 19,352,048 floats = ~77.4 MB

// ---------------------------------------------------------------------------
// K0: tiny precompute of all rank-1 collapse vectors (1 block, 512 threads)
// ---------------------------------------------------------------------------
__global__ void k_pre(const float* __restrict__ emW, const float* __restrict__ emb,
                      const float* __restrict__ lin1W, const float* __restrict__ lin1b,
                      const float* __restrict__ nn1W, const float* __restrict__ nn1b,
                      const float* __restrict__ Wih, const float* __restrict__ bih,
                      const float* __restrict__ bhh,
                      const float* __restrict__ lin3W, const float* __restrict__ lin3b,
                      float* __restrict__ ws) {
  int t = threadIdx.x;  // 0..511
  const float* wr = Wih + t * H;
  float su = 0.f, sv = 0.f;
  for (int k = 0; k < H; ++k) { su += wr[k] * nn1W[k]; sv += wr[k] * nn1b[k]; }
  ws[OFF_U + t] = su;
  ws[OFF_V + t] = sv + bih[t] + bhh[t];
  if (t < H) {
    const float* lr = lin3W + t * H;
    float sp = 0.f, sq = 0.f;
    for (int k = 0; k < H; ++k) { sp += lr[k] * emW[k]; sq += lr[k] * emb[k]; }
    ws[OFF_P + t] = sp;
    ws[OFF_Q + t] = sq + lin3b[t];
  }
  if (t == 0) {
    float a = 0.f, b = 0.f;
    for (int k = 0; k < H; ++k) { a += emW[k] * lin1W[k]; b += emb[k] * lin1W[k]; }
    ws[OFF_SC + 0] = a;
    ws[OFF_SC + 1] = b + lin1b[0];
  }
}

// ---------------------------------------------------------------------------
// K1: conv1 scatter — scalar messages, atomic segment-sum by dst
// ---------------------------------------------------------------------------
__global__ void k_conv1(const float* __restrict__ x, const float* __restrict__ ea,
                        const int* __restrict__ src, const int* __restrict__ dst,
                        float* __restrict__ ws) {
  int e = blockIdx.x * 256 + threadIdx.x;
  if (e >= N_EDGES) return;
  float alpha = ws[OFF_SC], beta = ws[OFF_SC + 1];
  float m = x[src[e]] + alpha * ea[e] + beta;
  m = m > 0.f ? m : 0.f;
  atomicAdd(&ws[OFF_S + dst[e]], m);
}

// ---------------------------------------------------------------------------
// K2: fused conv1-nn + LSTM step (rank-1 gates), one thread per (node, ch)
// ---------------------------------------------------------------------------
__global__ void k_lstm(const float* __restrict__ x, float* __restrict__ ws) {
  int i = blockIdx.x * 256 + threadIdx.x;  // 0 .. N*H-1
  int node = i >> 7;
  int k = i & (H - 1);
  float s = x[node] + ws[OFF_S + node];
  float ig = s * ws[OFF_U + k]        + ws[OFF_V + k];         // input gate
  float gg = s * ws[OFF_U + 256 + k]  + ws[OFF_V + 256 + k];   // cell gate
  float og = s * ws[OFF_U + 384 + k]  + ws[OFF_V + 384 + k];   // output gate
  float c = (1.f / (1.f + expf(-ig))) * tanhf(gg);
  float h = (1.f / (1.f + expf(-og))) * tanhf(c);
  ws[OFF_H2 + (size_t)i] = h;
}

// ---------------------------------------------------------------------------
// K3: conv3 scatter — 128-wide messages, atomic segment-sum. 2 edges / block.
// ---------------------------------------------------------------------------
__global__ void k_conv3(const float* __restrict__ ea, const int* __restrict__ src,
                        const int* __restrict__ dst, float* __restrict__ ws) {
  int e = blockIdx.x * 2 + (threadIdx.x >> 7);
  int k = threadIdx.x & (H - 1);
  float w = ea[e];
  int s = src[e], d = dst[e];
  float m = ws[OFF_H2 + (size_t)s * H + k] + w * ws[OFF_P + k] + ws[OFF_Q + k];
  m = m > 0.f ? m : 0.f;
  atomicAdd(&ws[OFF_AG + (size_t)d * H + k], m);
}

// ---------------------------------------------------------------------------
// K4: conv3 nn GEMM via V_WMMA_F32_16X16X4_F32 (fp32 in / fp32 out).
//     h3[16 x 128] = (h2 + aggr3)[16 x 128] @ nn3_W.T + nn3_b
//     block = 256 threads = 8 waves; wave w owns output columns [16w, 16w+16).
//     A (16x4 f32): lanes 0-15 -> M=lane, K={0,1}; lanes 16-31 -> K={2,3}.
//     B (4x16 f32): lanes 0-15 -> N=lane, K rows {0,1}; lanes 16-31 -> {2,3}.
//     D: VGPR j -> M=j (lanes 0-15) / M=8+j (lanes 16-31), N = lane%16.
// ---------------------------------------------------------------------------
__global__ void k_gemm(const float* __restrict__ nn3W, const float* __restrict__ nn3b,
                       float* __restrict__ ws) {
  __shared__ float zt[16 * 132];  // 16 rows, padded stride 132 (bank-conflict free)
  const float* h2 = ws + OFF_H2;
  const float* ag = ws + OFF_AG;
  float*       h3 = ws + OFF_H3;
  int node0 = blockIdx.x * 16;
  int t = threadIdx.x;

  // cooperative load: z = h2 + aggr3, 16x128 floats as 512 float4 chunks
  for (int i = t; i < 512; i += 256) {
    int row = i >> 5, cv = i & 31;
    const float4* hv = (const float4*)(h2 + (size_t)(node0 + row) * H);
    const float4* av = (const float4*)(ag + (size_t)(node0 + row) * H);
    float4 a = hv[cv], b = av[cv];
    float4 r; r.x = a.x + b.x; r.y = a.y + b.y; r.z = a.z + b.z; r.w = a.w + b.w;
    *(float4*)(zt + row * 132 + cv * 4) = r;
  }
  __syncthreads();

  int wave = t >> 5, lane = t & 31;
  int lm = lane & 15, hi = lane >> 4;
  int c0 = wave * 16;

  // accumulator init with bias (bias depends only on output column)
  float bv = nn3b[c0 + lm];
  v8f acc;
#pragma unroll
  for (int j = 0; j < 8; ++j) acc[j] = bv;

  const float* wrow = nn3W + (size_t)(c0 + lm) * H;  // B[k][n] = nn3_W[c0+n][k]
  const float* zrow = zt + lm * 132;                 // A[m][k] = z[m][k]
#pragma unroll 8
  for (int kk = 0; kk < 32; ++kk) {
    int kb = kk * 4 + 2 * hi;
    v2f a = *(const v2f*)(zrow + kb);
    v2f b = *(const v2f*)(wrow + kb);
    acc = __builtin_amdgcn_wmma_f32_16x16x4_f32(false, a, false, b,
                                                (short)0, acc, false, false);
  }

  // store D tile
  float* dbase = h3 + (size_t)(node0 + hi * 8) * H + (c0 + lm);
#pragma unroll
  for (int j = 0; j < 8; ++j) dbase[(size_t)j * H] = acc[j];
}

// ---------------------------------------------------------------------------
// K5: per-node decode dots: a[i] = h3[i].dec_W[:128], b[i] = h3[i].dec_W[128:]
//     one wave per node, __shfl_xor reduction (wave32)
// ---------------------------------------------------------------------------
__global__ void k_nodedot(const float* __restrict__ decW, float* __restrict__ ws) {
  int wave = threadIdx.x >> 5, lane = threadIdx.x & 31;
  int node = blockIdx.x * 8 + wave;
  const float* row = ws + OFF_H3 + (size_t)node * H;
  float a = 0.f, b = 0.f;
  for (int k = lane; k < H; k += 32) {
    float h = row[k];
    a += h * decW[k];
    b += h * decW[H + k];
  }
  for (int m = 16; m > 0; m >>= 1) {
    a += __shfl_xor(a, m, 32);
    b += __shfl_xor(b, m, 32);
  }
  if (lane == 0) { ws[OFF_A + node] = a; ws[OFF_B + node] = b; }
}

// ---------------------------------------------------------------------------
// K6: per-edge output gather
// ---------------------------------------------------------------------------
__global__ void k_out(const int* __restrict__ src, const int* __restrict__ dst,
                      const float* __restrict__ decb, const float* __restrict__ ws,
                      float* __restrict__ out) {
  int e = blockIdx.x * 256 + threadIdx.x;
  if (e >= N_EDGES) return;
  out[e] = ws[OFF_A + src[e]] + ws[OFF_B + dst[e]] + decb[0];
}

// ---------------------------------------------------------------------------
extern "C" void kernel_launch(void* const* d_in, const int* in_sizes, int n_in,
                              void* d_out, int out_size, void* d_ws, size_t ws_size,
                              hipStream_t stream) {
  const float* x     = (const float*)d_in[0];
  const float* ea    = (const float*)d_in[1];
  const int*   ei    = (const int*)  d_in[2];
  const int*   src   = ei;
  const int*   dst   = ei + N_EDGES;
  const float* emW   = (const float*)d_in[3];
  const float* emb   = (const float*)d_in[4];
  const float* lin1W = (const float*)d_in[5];
  const float* lin1b = (const float*)d_in[6];
  const float* nn1W  = (const float*)d_in[7];
  const float* nn1b  = (const float*)d_in[8];
  const float* Wih   = (const float*)d_in[9];
  // d_in[10] = W_hh : provably unused (zero initial hidden state)
  const float* bih   = (const float*)d_in[11];
  const float* bhh   = (const float*)d_in[12];
  const float* lin3W = (const float*)d_in[13];
  const float* lin3b = (const float*)d_in[14];
  const float* nn3W  = (const float*)d_in[15];
  const float* nn3b  = (const float*)d_in[16];
  const float* decW  = (const float*)d_in[17];
  const float* decb  = (const float*)d_in[18];
  float* out = (float*)d_out;
  float* ws  = (float*)d_ws;

  // zero the two atomic accumulators (graph-capture-safe async memset)
  hipMemsetAsync(ws + OFF_S,  0, (size_t)N_NODES * sizeof(float), stream);
  hipMemsetAsync(ws + OFF_AG, 0, (size_t)N_NODES * H * sizeof(float), stream);

  k_pre<<<1, 512, 0, stream>>>(emW, emb, lin1W, lin1b, nn1W, nn1b,
                               Wih, bih, bhh, lin3W, lin3b, ws);

  k_conv1<<<(N_EDGES + 255) / 256, 256, 0, stream>>>(x, ea, src, dst, ws);

  k_lstm<<<(N_NODES * H) / 256, 256, 0, stream>>>(x, ws);   // 25000 blocks exact

  k_conv3<<<N_EDGES / 2, 256, 0, stream>>>(ea, src, dst, ws);

  k_gemm<<<N_NODES / 16, 256, 0, stream>>>(nn3W, nn3b, ws); // 3125 blocks exact

  k_nodedot<<<N_NODES / 8, 256, 0, stream>>>(decW, ws);     // 6250 blocks exact

  k_out<<<(N_EDGES + 255) / 256, 256, 0, stream>>>(src, dst, decb, ws, out);
}